// PointConv_10814727651412
// MI455X (gfx1250) — compile-verified
//
#include <hip/hip_runtime.h>
#include <hip/hip_bf16.h>

typedef __attribute__((ext_vector_type(2))) float v2f;
typedef __attribute__((ext_vector_type(8))) float v8f;

#define BB   8
#define NN   8192
#define SS   2048
#define KK   32
#define CIN  64
#define COUT 128
#define BSZ  (BB * SS)              /* 16384 */
#define MBIG ((size_t)BSZ * KK)     /* 524288 */

#if defined(__gfx1250__) && __has_builtin(__builtin_amdgcn_global_load_async_to_lds_b128) && \
    __has_builtin(__builtin_amdgcn_s_wait_asynccnt)
#define USE_ASYNC_LDS 1
typedef int v4i_vs __attribute__((vector_size(4 * sizeof(int))));
typedef __attribute__((address_space(1))) v4i_vs* glb_v4i_p;
typedef __attribute__((address_space(3))) v4i_vs* lds_v4i_p;
#else
#define USE_ASYNC_LDS 0
#endif

// ---------------------------------------------------------------------------
// 0. zero a float region (stats accumulators — must be re-zeroed every call)
// ---------------------------------------------------------------------------
__global__ void zero_kernel(float* p, int n) {
    int i = blockIdx.x * blockDim.x + threadIdx.x;
    if (i < n) p[i] = 0.0f;
}

// ---------------------------------------------------------------------------
// 1. Furthest point sampling: one block per batch, dist[] lives in LDS.
// ---------------------------------------------------------------------------
__global__ __launch_bounds__(256) void fps_kernel(const float* __restrict__ xyz,
                                                  int* __restrict__ fidx) {
    const int b = blockIdx.x;
    const float* P = xyz + (size_t)b * NN * 3;
    __shared__ float dist[NN];
    __shared__ float rv[256];
    __shared__ int   ri[256];
    __shared__ int   s_far;
    for (int i = threadIdx.x; i < NN; i += 256) dist[i] = 1e10f;
    if (threadIdx.x == 0) s_far = 0;
    __syncthreads();
    for (int t = 0; t < SS; ++t) {
        const int far = s_far;
        if (threadIdx.x == 0) fidx[b * SS + t] = far;
        const float cx = P[far * 3 + 0];
        const float cy = P[far * 3 + 1];
        const float cz = P[far * 3 + 2];
        float bv = -1.0f; int bi = 0;
        for (int i = threadIdx.x; i < NN; i += 256) {
            const float dx = P[i * 3 + 0] - cx;
            const float dy = P[i * 3 + 1] - cy;
            const float dz = P[i * 3 + 2] - cz;
            float d = dx * dx + dy * dy + dz * dz;
            d = fminf(dist[i], d);
            dist[i] = d;
            if (d > bv) { bv = d; bi = i; }
        }
        rv[threadIdx.x] = bv; ri[threadIdx.x] = bi;
        __syncthreads();
        for (int s = 128; s > 0; s >>= 1) {
            if (threadIdx.x < s) {
                const float ov = rv[threadIdx.x + s];
                const int   oi = ri[threadIdx.x + s];
                if (ov > rv[threadIdx.x] ||
                    (ov == rv[threadIdx.x] && oi < ri[threadIdx.x])) {
                    rv[threadIdx.x] = ov; ri[threadIdx.x] = oi;
                }
            }
            __syncthreads();
        }
        if (threadIdx.x == 0) s_far = ri[0];
        __syncthreads();
    }
}

// ---------------------------------------------------------------------------
// 2. Gather sampled xyz (-> d_out) and repack sampled features to [CIN][B*S]
// ---------------------------------------------------------------------------
__global__ void gather_kernel(const float* __restrict__ xyz,
                              const float* __restrict__ feat,
                              const int* __restrict__ fidx,
                              float* __restrict__ out_xyz,
                              float* __restrict__ sf) {
    const int gs = blockIdx.x * blockDim.x + threadIdx.x;   // over B*S
    if (gs >= BSZ) return;
    const int b = gs / SS;
    const int id = fidx[gs];
    const float* p = xyz + ((size_t)b * NN + id) * 3;
    float* o = out_xyz + (size_t)gs * 3;
    o[0] = p[0]; o[1] = p[1]; o[2] = p[2];
    const float* fb = feat + (size_t)b * CIN * NN + id;
    for (int i = 0; i < CIN; ++i)
        sf[(size_t)i * BSZ + gs] = fb[(size_t)i * NN];
}

// ---------------------------------------------------------------------------
// 3. KNN: one thread per query, candidate points tiled through LDS
//    (async global->LDS b128 staging when the toolchain exposes the builtin).
// ---------------------------------------------------------------------------
__global__ __launch_bounds__(256) void knn_kernel(const float* __restrict__ xyz,
                                                  const float* __restrict__ qxyz,
                                                  int* __restrict__ knn) {
    const int b = blockIdx.y;
    const int s = blockIdx.x * 256 + threadIdx.x;
    const float* P = xyz + (size_t)b * NN * 3;
    const float* q = qxyz + ((size_t)b * SS + s) * 3;
    const float qx = q[0], qy = q[1], qz = q[2];

    float d[KK]; int id[KK];
#pragma unroll
    for (int k = 0; k < KK; ++k) { d[k] = 3.4e38f; id[k] = 0; }
    float worst = 3.4e38f; int wpos = 0;

    __shared__ __align__(16) float tile[3 * 1024];   // x,y,z interleaved per point
    for (int base = 0; base < NN; base += 1024) {
        __syncthreads();
#if USE_ASYNC_LDS
        for (int j = threadIdx.x; j < 768; j += 256) {   // 3072 floats / 4-wide
            glb_v4i_p gp = (glb_v4i_p)(P + (size_t)base * 3 + j * 4);
            lds_v4i_p lp = (lds_v4i_p)&tile[j * 4];
            __builtin_amdgcn_global_load_async_to_lds_b128(gp, lp, 0, 0);
        }
        __builtin_amdgcn_s_wait_asynccnt(0);
#else
        for (int j = threadIdx.x; j < 3072; j += 256)
            tile[j] = P[(size_t)base * 3 + j];
#endif
        __syncthreads();
        for (int j = 0; j < 1024; ++j) {
            const float dx = tile[j * 3 + 0] - qx;
            const float dy = tile[j * 3 + 1] - qy;
            const float dz = tile[j * 3 + 2] - qz;
            const float dd = dx * dx + dy * dy + dz * dz;
            if (dd < worst) {
#pragma unroll
                for (int k = 0; k < KK; ++k)
                    if (k == wpos) { d[k] = dd; id[k] = base + j; }
                worst = d[0]; wpos = 0;
#pragma unroll
                for (int k = 1; k < KK; ++k)
                    if (d[k] > worst) { worst = d[k]; wpos = k; }
            }
        }
    }
    // move nearest to slot 0 (anchor for prenorm)
    float best = d[0]; int bpos = 0;
#pragma unroll
    for (int k = 1; k < KK; ++k) if (d[k] < best) { best = d[k]; bpos = k; }
    int tmp;
#pragma unroll
    for (int k = 0; k < KK; ++k)
        if (k == bpos) { tmp = id[0]; id[0] = id[k]; id[k] = tmp; }
    int* out = knn + ((size_t)b * SS + s) * KK;
#pragma unroll
    for (int k = 0; k < KK; ++k) out[k] = id[k];
}

// ---------------------------------------------------------------------------
// 4. Gather grouped features + prenorm -> G [CIN][M] (M = B*S*K)
// ---------------------------------------------------------------------------
__global__ __launch_bounds__(256) void prenorm_kernel(const float* __restrict__ feat,
                                                      const int* __restrict__ knn,
                                                      const float* __restrict__ alpha,
                                                      const float* __restrict__ beta,
                                                      float* __restrict__ g) {
    const size_t m = (size_t)blockIdx.x * blockDim.x + threadIdx.x;   // over B*S*K
    if (m >= MBIG) return;
    const size_t bs = m >> 5;
    const int b = (int)(bs / SS);
    const int idv = knn[m];
    const int id0 = knn[bs << 5];
    const float* fb = feat + (size_t)b * CIN * NN;
    float c[CIN];
    float sum = 0.0f;
#pragma unroll
    for (int i = 0; i < CIN; ++i) {
        const float v = fb[(size_t)i * NN + idv];
        const float a = fb[(size_t)i * NN + id0];
        c[i] = v - a; sum += c[i];
    }
    const float mean = sum * (1.0f / CIN);
    float q = 0.0f;
#pragma unroll
    for (int i = 0; i < CIN; ++i) { const float t = c[i] - mean; q += t * t; }
    const float stdv = sqrtf(q * (1.0f / CIN));
    const float inv = 1.0f / (stdv + 1e-6f);
#pragma unroll
    for (int i = 0; i < CIN; ++i)
        g[(size_t)i * MBIG + m] = c[i] * inv * alpha[i] + beta[i];
}

// ---------------------------------------------------------------------------
// 5. fp32 WMMA GEMM:  Y[Co][M] = W[Co][Ci] * affine(X[Ci][M]) + bias
//    Compile-time AFFINE/RELU specialization keeps the K-loop branch-free.
//    Per-wave: 16 rows x 64 cols (4 accumulator tiles sharing one A fragment).
// ---------------------------------------------------------------------------
template <int AFFINE, int RELU>
__global__ __launch_bounds__(256) void wmma_gemm_kernel(
    const float* __restrict__ W, const float* __restrict__ bias,
    const float* __restrict__ X, const float* __restrict__ scaleIn,
    const float* __restrict__ shiftIn,
    float* __restrict__ Y, int M, int Ci) {
    const int lane = threadIdx.x & 31;
    const int wave = threadIdx.x >> 5;
    const int lm = lane & 15;
    const int hi = lane >> 4;
    const int o0 = blockIdx.y << 4;
    const int m0 = blockIdx.x * 512 + wave * 64;

    v8f acc[4] = {};
    const float* wrow = W + (size_t)(o0 + lm) * Ci;

#pragma unroll 2
    for (int k = 0; k < Ci; k += 4) {
        const int k0 = k + (hi << 1);
        v2f a;
        a.x = wrow[k0];
        a.y = wrow[k0 + 1];
        float s0, t0, s1, t1;
        if (AFFINE) {
            s0 = scaleIn[k0];     t0 = shiftIn[k0];
            s1 = scaleIn[k0 + 1]; t1 = shiftIn[k0 + 1];
        }
        const float* xp0 = X + (size_t)k0 * M + m0 + lm;
        const float* xp1 = xp0 + M;
        if (k + 4 < Ci) __builtin_prefetch(xp0 + (size_t)4 * M, 0, 1);
#pragma unroll
        for (int t = 0; t < 4; ++t) {
            float x0 = xp0[t * 16];
            float x1 = xp1[t * 16];
            if (AFFINE) { x0 = fmaf(x0, s0, t0); x1 = fmaf(x1, s1, t1); }
            if (RELU)   { x0 = fmaxf(x0, 0.0f);  x1 = fmaxf(x1, 0.0f); }
            v2f bf; bf.x = x0; bf.y = x1;
            acc[t] = __builtin_amdgcn_wmma_f32_16x16x4_f32(
                false, a, false, bf, (short)0, acc[t], false, false);
        }
    }
#pragma unroll
    for (int r = 0; r < 8; ++r) {
        const int o = o0 + r + (hi << 3);
        const float bb = bias[o];
        float* yp = Y + (size_t)o * M + m0 + lm;
#pragma unroll
        for (int t = 0; t < 4; ++t) yp[t * 16] = acc[t][r] + bb;
    }
}

// ---------------------------------------------------------------------------
// 6. per-channel sum / sumsq over X[C][M] (hierarchical: LDS tree + atomics)
// ---------------------------------------------------------------------------
__global__ __launch_bounds__(256) void channel_stats_kernel(
    const float* __restrict__ X, int M, int chunk,
    float* __restrict__ sums, float* __restrict__ sumsq) {
    const int c = blockIdx.y;
    const size_t base = (size_t)c * M + (size_t)blockIdx.x * chunk;
    float s = 0.0f, q = 0.0f;
    for (int i = threadIdx.x; i < chunk; i += 256) {
        const float v = X[base + i];
        s += v; q += v * v;
    }
    __shared__ float rs[256], rq[256];
    rs[threadIdx.x] = s; rq[threadIdx.x] = q;
    __syncthreads();
    for (int st = 128; st > 0; st >>= 1) {
        if (threadIdx.x < st) {
            rs[threadIdx.x] += rs[threadIdx.x + st];
            rq[threadIdx.x] += rq[threadIdx.x + st];
        }
        __syncthreads();
    }
    if (threadIdx.x == 0) {
        atomicAdd(&sums[c], rs[0]);
        atomicAdd(&sumsq[c], rq[0]);
    }
}

// ---------------------------------------------------------------------------
// 7. fold BN stats + (gamma, beta) into per-channel (scale, shift)
// ---------------------------------------------------------------------------
__global__ void bn_finalize_kernel(const float* __restrict__ sums,
                                   const float* __restrict__ sumsq,
                                   const float* __restrict__ gamma,
                                   const float* __restrict__ beta,
                                   float invM,
                                   float* __restrict__ scale,
                                   float* __restrict__ shift) {
    const int c = threadIdx.x;
    if (c < COUT) {
        const float m = sums[c] * invM;
        const float v = sumsq[c] * invM - m * m;
        const float sc = gamma[c] * rsqrtf(v + 1e-5f);
        scale[c] = sc;
        shift[c] = beta[c] - m * sc;
    }
}

// ---------------------------------------------------------------------------
// 8. f = relu( max_k BN1(h1) + max_k BN0(h0) )  -> F [COUT][B*S]
// ---------------------------------------------------------------------------
__global__ void max_fuse_kernel(const float* __restrict__ h0,
                                const float* __restrict__ h1,
                                const float* __restrict__ sc0, const float* __restrict__ sh0,
                                const float* __restrict__ sc1, const float* __restrict__ sh1,
                                float* __restrict__ F) {
    const int t = blockIdx.x * blockDim.x + threadIdx.x;    // over COUT*B*S
    if (t >= COUT * BSZ) return;
    const int o = t / BSZ;
    const int bs = t - o * BSZ;
    const size_t m0 = (size_t)o * MBIG + (size_t)bs * KK;
    const float4* p0 = (const float4*)(h0 + m0);
    const float4* p1 = (const float4*)(h1 + m0);
    const float s0 = sc0[o], t0 = sh0[o], s1 = sc1[o], t1 = sh1[o];
    float a0 = -3.4e38f, a1 = -3.4e38f;
#pragma unroll
    for (int k = 0; k < KK / 4; ++k) {
        const float4 v0 = p0[k];
        const float4 v1 = p1[k];
        a0 = fmaxf(a0, fmaxf(fmaxf(fmaf(v0.x, s0, t0), fmaf(v0.y, s0, t0)),
                             fmaxf(fmaf(v0.z, s0, t0), fmaf(v0.w, s0, t0))));
        a1 = fmaxf(a1, fmaxf(fmaxf(fmaf(v1.x, s1, t1), fmaf(v1.y, s1, t1)),
                             fmaxf(fmaf(v1.z, s1, t1), fmaf(v1.w, s1, t1))));
    }
    F[t] = fmaxf(a1 + a0, 0.0f);
}

// ---------------------------------------------------------------------------
// 9. new_features = relu( BN3(y3) + BNsk(skip) )   layout -> [B][COUT][S]
// ---------------------------------------------------------------------------
__global__ void final_kernel(const float* __restrict__ y3,
                             const float* __restrict__ skp,
                             const float* __restrict__ sc3, const float* __restrict__ sh3,
                             const float* __restrict__ sck, const float* __restrict__ shk,
                             float* __restrict__ out) {
    const int t = blockIdx.x * blockDim.x + threadIdx.x;    // over COUT*B*S
    if (t >= COUT * BSZ) return;
    const int o = t / BSZ;
    const int bs = t - o * BSZ;
    const int b = bs / SS, s = bs - b * SS;
    const float v = fmaf(y3[t], sc3[o], sh3[o]) + fmaf(skp[t], sck[o], shk[o]);
    out[((size_t)b * COUT + o) * SS + s] = fmaxf(v, 0.0f);
}

// ---------------------------------------------------------------------------
extern "C" void kernel_launch(void* const* d_in, const int* in_sizes, int n_in,
                              void* d_out, int out_size, void* d_ws, size_t ws_size,
                              hipStream_t stream) {
    const float* xyz      = (const float*)d_in[0];
    const float* features = (const float*)d_in[1];
    const float* alpha    = (const float*)d_in[2];
    const float* beta_pn  = (const float*)d_in[3];
    const float* W0 = (const float*)d_in[4];  const float* b0  = (const float*)d_in[5];
    const float* g0 = (const float*)d_in[6];  const float* be0 = (const float*)d_in[7];
    const float* W1 = (const float*)d_in[8];  const float* b1  = (const float*)d_in[9];
    const float* g1 = (const float*)d_in[10]; const float* be1 = (const float*)d_in[11];
    const float* W2 = (const float*)d_in[12]; const float* b2  = (const float*)d_in[13];
    const float* g2 = (const float*)d_in[14]; const float* be2 = (const float*)d_in[15];
    const float* W3 = (const float*)d_in[16]; const float* b3  = (const float*)d_in[17];
    const float* g3 = (const float*)d_in[18]; const float* be3 = (const float*)d_in[19];
    const float* Wsk = (const float*)d_in[20]; const float* bsk  = (const float*)d_in[21];
    const float* gsk = (const float*)d_in[22]; const float* besk = (const float*)d_in[23];

    float* out_xyz  = (float*)d_out;                       // [B*S][3]
    float* out_feat = out_xyz + (size_t)BSZ * 3;           // [B][COUT][S]

    char* w = (char*)d_ws;
    int*   fidx = (int*)w;    w += sizeof(int) * (size_t)BSZ;
    int*   knn  = (int*)w;    w += sizeof(int) * MBIG;
    float* sf   = (float*)w;  w += sizeof(float) * (size_t)CIN  * BSZ;
    float* gmat = (float*)w;  w += sizeof(float) * (size_t)CIN  * MBIG;
    float* h0   = (float*)w;  w += sizeof(float) * (size_t)COUT * MBIG;
    float* h1   = (float*)w;  w += sizeof(float) * (size_t)COUT * MBIG;
    float* F    = (float*)w;  w += sizeof(float) * (size_t)COUT * BSZ;
    float* y2   = (float*)w;  w += sizeof(float) * (size_t)COUT * BSZ;
    float* y3   = (float*)w;  w += sizeof(float) * (size_t)COUT * BSZ;
    float* skp  = (float*)w;  w += sizeof(float) * (size_t)COUT * BSZ;
    float* stats = (float*)w;                // 5 sets x [sums|sumsq] x 128
    float* affine = stats + 5 * 256;         // 5 sets x [scale|shift] x 128
#define SUMS(i)  (stats + (i) * 256)
#define SUMQ(i)  (stats + (i) * 256 + 128)
#define SCL(i)   (affine + (i) * 256)
#define SHF(i)   (affine + (i) * 256 + 128)

    // stats accumulators must be zero before atomics
    zero_kernel<<<(5 * 256 + 255) / 256, 256, 0, stream>>>(stats, 5 * 256);

    // FPS + gathers + KNN + prenorm
    fps_kernel<<<BB, 256, 0, stream>>>(xyz, fidx);
    gather_kernel<<<(BSZ + 255) / 256, 256, 0, stream>>>(xyz, features, fidx, out_xyz, sf);
    knn_kernel<<<dim3(SS / 256, BB), 256, 0, stream>>>(xyz, out_xyz, knn);
    prenorm_kernel<<<(int)(MBIG / 256), 256, 0, stream>>>(features, knn, alpha, beta_pn, gmat);

    // conv0 (WMMA) -> h0 ; BN0 stats
    wmma_gemm_kernel<0, 0><<<dim3((int)(MBIG / 512), COUT / 16), 256, 0, stream>>>(
        W0, b0, gmat, nullptr, nullptr, h0, (int)MBIG, CIN);
    channel_stats_kernel<<<dim3((int)(MBIG / 8192), COUT), 256, 0, stream>>>(
        h0, (int)MBIG, 8192, SUMS(0), SUMQ(0));
    bn_finalize_kernel<<<1, 128, 0, stream>>>(SUMS(0), SUMQ(0), g0, be0,
                                              1.0f / (float)MBIG, SCL(0), SHF(0));

    // conv1 (WMMA, BN0 folded into operand load) -> h1 ; BN1 stats
    wmma_gemm_kernel<1, 0><<<dim3((int)(MBIG / 512), COUT / 16), 256, 0, stream>>>(
        W1, b1, h0, SCL(0), SHF(0), h1, (int)MBIG, COUT);
    channel_stats_kernel<<<dim3((int)(MBIG / 8192), COUT), 256, 0, stream>>>(
        h1, (int)MBIG, 8192, SUMS(1), SUMQ(1));
    bn_finalize_kernel<<<1, 128, 0, stream>>>(SUMS(1), SUMQ(1), g1, be1,
                                              1.0f / (float)MBIG, SCL(1), SHF(1));

    // f = relu(max_k BN1(h1) + max_k BN0(h0))
    max_fuse_kernel<<<(COUT * BSZ + 255) / 256, 256, 0, stream>>>(
        h0, h1, SCL(0), SHF(0), SCL(1), SHF(1), F);

    // conv2 (WMMA) -> y2 ; BN2 stats
    wmma_gemm_kernel<0, 0><<<dim3(BSZ / 512, COUT / 16), 256, 0, stream>>>(
        W2, b2, F, nullptr, nullptr, y2, BSZ, COUT);
    channel_stats_kernel<<<dim3(BSZ / 8192, COUT), 256, 0, stream>>>(
        y2, BSZ, 8192, SUMS(2), SUMQ(2));
    bn_finalize_kernel<<<1, 128, 0, stream>>>(SUMS(2), SUMQ(2), g2, be2,
                                              1.0f / (float)BSZ, SCL(2), SHF(2));

    // conv3 (WMMA, relu(BN2) folded into operand load) -> y3 ; BN3 stats
    wmma_gemm_kernel<1, 1><<<dim3(BSZ / 512, COUT / 16), 256, 0, stream>>>(
        W3, b3, y2, SCL(2), SHF(2), y3, BSZ, COUT);
    channel_stats_kernel<<<dim3(BSZ / 8192, COUT), 256, 0, stream>>>(
        y3, BSZ, 8192, SUMS(3), SUMQ(3));
    bn_finalize_kernel<<<1, 128, 0, stream>>>(SUMS(3), SUMQ(3), g3, be3,
                                              1.0f / (float)BSZ, SCL(3), SHF(3));

    // skip conv (WMMA) -> skp ; BNsk stats
    wmma_gemm_kernel<0, 0><<<dim3(BSZ / 512, COUT / 16), 256, 0, stream>>>(
        Wsk, bsk, sf, nullptr, nullptr, skp, BSZ, CIN);
    channel_stats_kernel<<<dim3(BSZ / 8192, COUT), 256, 0, stream>>>(
        skp, BSZ, 8192, SUMS(4), SUMQ(4));
    bn_finalize_kernel<<<1, 128, 0, stream>>>(SUMS(4), SUMQ(4), gsk, besk,
                                              1.0f / (float)BSZ, SCL(4), SHF(4));

    // new_features = relu(BN3(y3) + BNsk(skip))
    final_kernel<<<(COUT * BSZ + 255) / 256, 256, 0, stream>>>(
        y3, skp, SCL(3), SHF(3), SCL(4), SHF(4), out_feat);
#undef SUMS
#undef SUMQ
#undef SCL
#undef SHF
}